// MahalanobisDistance_17660905521624
// MI455X (gfx1250) — compile-verified
//
#include <hip/hip_runtime.h>
#include <hip/hip_bf16.h>
#include <math.h>

// Shapes: P=128, S=256, EP=1, ES=64, D=64, H=512
typedef __attribute__((ext_vector_type(2))) float v2f;
typedef __attribute__((ext_vector_type(8))) float v8f;

// ---------------------------------------------------------------------------
// Kernel 1: L2-normalize rows of length 64.  wave-per-row, lane holds float2.
// ---------------------------------------------------------------------------
__global__ __launch_bounds__(256) void norm_rows64(const float* __restrict__ src,
                                                   float* __restrict__ dst,
                                                   int nrows) {
  int w = blockIdx.x * 8 + (threadIdx.x >> 5);
  int lane = threadIdx.x & 31;
  if (w >= nrows) return;
  const float2* s = (const float2*)(src + (size_t)w * 64);
  float2 v = s[lane];
  float ss = v.x * v.x + v.y * v.y;
#pragma unroll
  for (int off = 16; off > 0; off >>= 1) ss += __shfl_xor(ss, off, 32);
  float inv = 1.0f / fmaxf(sqrtf(ss), 1e-12f);
  float2 o;
  o.x = v.x * inv;
  o.y = v.y * inv;
  ((float2*)(dst + (size_t)w * 64))[lane] = o;
}

// ---------------------------------------------------------------------------
// Kernel 2: H = relu(satN [16384,64] @ W1^T [64,512] + b1) -> [16384,512]
// One 16x16 output tile per wave, K=64 chained as 16 x WMMA_F32_16X16X4_F32.
// ---------------------------------------------------------------------------
__global__ __launch_bounds__(256) void mlp1(const float* __restrict__ satN,
                                            const float* __restrict__ W1,
                                            const float* __restrict__ b1,
                                            float* __restrict__ Hout) {
  int wave = blockIdx.x * 8 + (threadIdx.x >> 5);  // 0..32767
  int lane = threadIdx.x & 31;
  int l15 = lane & 15, half = lane >> 4;
  int mt = wave >> 5;  // 0..1023 (rows of 16 (s,e) pairs)
  int nt = wave & 31;  // 0..31   (cols of 16 h)
  int row0 = mt * 16;
  int col = nt * 16 + l15;
  // A[m,k]: a[i] = satN[row0+l15][4kk + 2*half + i]
  const v2f* aP = (const v2f*)(satN + (size_t)(row0 + l15) * 64 + 2 * half);
  // B[k,n] = W1[n,k]: b[i] = W1[col][4kk + 2*half + i]
  const v2f* bP = (const v2f*)(W1 + (size_t)col * 64 + 2 * half);
  v8f acc = {};
#pragma unroll
  for (int kk = 0; kk < 16; ++kk) {
    v2f a = aP[2 * kk];
    v2f b = bP[2 * kk];
    acc = __builtin_amdgcn_wmma_f32_16x16x4_f32(false, a, false, b, (short)0,
                                                acc, false, false);
  }
  float bias = b1[col];
#pragma unroll
  for (int v = 0; v < 8; ++v) {
    int r = row0 + v + 8 * half;
    Hout[(size_t)r * 512 + col] = fmaxf(acc[v] + bias, 0.0f);
  }
}

// ---------------------------------------------------------------------------
// Kernel 3: fused layer-2 GEMM + Mahalanobis application + max-over-e.
// One workgroup (256 thr, 8 waves) per s.  For each e-block of 16 (s,e) rows,
// the 4096-wide layer-2 output is produced in 16 chunks of 256 columns
// (= 4 complete k-rows of A[64,64] per (s,e)); each chunk is consumed
// immediately: Y = A'[64,64] @ panoN^T[64,128], d2 += (Y - A'satN)^2.
// out[p,s] = max_e d2.   No 268MB m_flat intermediate ever hits memory.
// ---------------------------------------------------------------------------
#define HS_STRIDE 516   // 512 + 4 pad  -> conflict-free A-frag reads
#define A2_STRIDE 68    // 64 + 4 pad
#define D2_STRIDE 132   // 128 + 4 pad

__global__ __launch_bounds__(256) void fused_tail(const float* __restrict__ Hbuf,  // [16384,512]
                                                  const float* __restrict__ satN,  // [16384,64]
                                                  const float* __restrict__ panoN, // [128,64]
                                                  const float* __restrict__ W2,    // [4096,512]
                                                  const float* __restrict__ b2,    // [4096]
                                                  float* __restrict__ out)         // [128,256]
{
  __shared__ float Hs[16 * HS_STRIDE];    // 33024 B : H rows for 16 (s,e)
  __shared__ float A2[64 * A2_STRIDE];    // 17408 B : A' rows (se*4+k) x f
  __shared__ float d2acc[16 * D2_STRIDE]; //  8448 B : d2 per (se, p)
  __shared__ float satB[16 * 64];         //  4096 B
  __shared__ float ys[64];                //   256 B : A' @ satN per row
  __shared__ float omax[128];             //   512 B : running max over e

  const int s = blockIdx.x;  // 0..255
  const int tid = threadIdx.x;
  const int wave = tid >> 5;
  const int lane = tid & 31;
  const int l15 = lane & 15, half = lane >> 4;

  if (tid < 128) omax[tid] = 0.0f;

  for (int eb = 0; eb < 4; ++eb) {
    const int seBase = s * 64 + eb * 16;
    __syncthreads();  // prior eb done with Hs / d2acc
    for (int i = tid; i < 16 * 512; i += 256) {
      int r = i >> 9, c = i & 511;
      Hs[r * HS_STRIDE + c] = Hbuf[(size_t)(seBase + r) * 512 + c];
    }
    for (int i = tid; i < 16 * 64; i += 256)
      satB[i] = satN[(size_t)seBase * 64 + i];
    for (int i = tid; i < 16 * D2_STRIDE; i += 256) d2acc[i] = 0.0f;
    __syncthreads();

    for (int ch = 0; ch < 16; ++ch) {
      const int obase = ch * 256;
      // ---- GEMM1: A_chunk[16 se, 256 o] = Hs @ W2[obase:obase+256, :]^T ----
      v8f acc1[2];
      {
        const v2f* aP = (const v2f*)(Hs + l15 * HS_STRIDE + 2 * half);
#pragma unroll
        for (int t = 0; t < 2; ++t) {
          const int o = obase + (wave + 8 * t) * 16 + l15;
          const v2f* bP = (const v2f*)(W2 + (size_t)o * 512 + 2 * half);
          v8f acc = {};
#pragma unroll 8
          for (int kk = 0; kk < 128; ++kk) {
            v2f a = aP[2 * kk];
            v2f b = bP[2 * kk];
            acc = __builtin_amdgcn_wmma_f32_16x16x4_f32(false, a, false, b,
                                                        (short)0, acc, false, false);
          }
          acc1[t] = acc;
        }
      }
      __syncthreads();  // previous chunk's GEMM2 finished reading A2
      // writeback into A'[row=se*4+k][f]  (+ b2)
#pragma unroll
      for (int t = 0; t < 2; ++t) {
        const int ol = (wave + 8 * t) * 16 + l15;  // 0..255 within chunk
        const float bb = b2[obase + ol];
        const int kl = ol >> 6, f = ol & 63;
#pragma unroll
        for (int v = 0; v < 8; ++v) {
          const int se = v + 8 * half;
          A2[(se * 4 + kl) * A2_STRIDE + f] = acc1[t][v] + bb;
        }
      }
      __syncthreads();
      // ---- ys[r] = sum_f A'[r,f] * satN[se,f]  (tiny GEMV, VALU) ----
      if (tid < 64) {
        const int se = tid >> 2;
        float a = 0.0f;
#pragma unroll 8
        for (int f = 0; f < 64; ++f)
          a += A2[tid * A2_STRIDE + f] * satB[se * 64 + f];
        ys[tid] = a;
      }
      __syncthreads();
      // ---- GEMM2: Y[64,128] = A'[64,64] @ panoN^T, fused square-accumulate --
      {
        const int p = wave * 16 + l15;  // wave owns one 16-wide p tile
        const v2f* bP = (const v2f*)(panoN + (size_t)p * 64 + 2 * half);
#pragma unroll
        for (int mt = 0; mt < 4; ++mt) {
          const int m0 = mt * 16;
          const v2f* aP = (const v2f*)(A2 + (m0 + l15) * A2_STRIDE + 2 * half);
          v8f acc = {};
#pragma unroll
          for (int kk = 0; kk < 16; ++kk) {
            v2f a = aP[2 * kk];
            v2f b = bP[2 * kk];
            acc = __builtin_amdgcn_wmma_f32_16x16x4_f32(false, a, false, b,
                                                        (short)0, acc, false, false);
          }
          const int r0 = m0 + 8 * half;  // first of 8 rows this lane holds
          float s0 = 0.0f, s1 = 0.0f;
#pragma unroll
          for (int v = 0; v < 4; ++v) {
            float d0 = acc[v] - ys[r0 + v];
            float d1 = acc[v + 4] - ys[r0 + 4 + v];
            s0 += d0 * d0;
            s1 += d1 * d1;
          }
          const int se0 = r0 >> 2;  // rows come 4-per-se; lane owns 2 se's
          d2acc[se0 * D2_STRIDE + p] += s0;        // race-free: unique (se,p)
          d2acc[(se0 + 1) * D2_STRIDE + p] += s1;  // per lane per chunk
        }
      }
      // next chunk's pre-writeback barrier protects A2; d2acc guarded below
    }
    __syncthreads();
    if (tid < 128) {  // fold 16 e's of this block into running max
      float m = omax[tid];
#pragma unroll
      for (int se = 0; se < 16; ++se)
        m = fmaxf(m, d2acc[se * D2_STRIDE + tid]);
      omax[tid] = m;
    }
  }
  __syncthreads();
  if (tid < 128) out[(size_t)tid * 256 + s] = omax[tid];  // out[p,s]
}

// ---------------------------------------------------------------------------
extern "C" void kernel_launch(void* const* d_in, const int* in_sizes, int n_in,
                              void* d_out, int out_size, void* d_ws, size_t ws_size,
                              hipStream_t stream) {
  (void)in_sizes; (void)n_in; (void)out_size; (void)ws_size;
  const float* sat  = (const float*)d_in[0];  // [256,64,64]
  const float* pano = (const float*)d_in[1];  // [128,1,64]
  const float* W1   = (const float*)d_in[2];  // [512,64]
  const float* b1   = (const float*)d_in[3];  // [512]
  const float* W2   = (const float*)d_in[4];  // [4096,512]
  const float* b2   = (const float*)d_in[5];  // [4096]
  float* out = (float*)d_out;                 // [128,256]

  char* ws = (char*)d_ws;
  float* satN  = (float*)ws;                            //  4 MB  [16384,64]
  float* panoN = (float*)(ws + 4194304);                // 32 KB  [128,64]
  float* Hbuf  = (float*)(ws + 4194304 + 32768);        // 32 MB  [16384,512]

  norm_rows64<<<16384 / 8, 256, 0, stream>>>(sat, satN, 16384);
  norm_rows64<<<128 / 8, 256, 0, stream>>>(pano, panoN, 128);
  mlp1<<<32768 / 8, 256, 0, stream>>>(satN, W1, b1, Hbuf);
  fused_tail<<<256, 256, 0, stream>>>(Hbuf, satN, panoN, W2, b2, out);
}